// IAGNNConv_36421322670667
// MI455X (gfx1250) — compile-verified
//
#include <hip/hip_runtime.h>
#include <hip/hip_bf16.h>
#include <math.h>

#define D 128

typedef __attribute__((ext_vector_type(2)))  float  v2f;
typedef __attribute__((ext_vector_type(8)))  float  v8f;

// ---------------- kernel 0: zero BN stat accumulators ----------------
__global__ void k_zero_stats(float* stats) {
    stats[threadIdx.x] = 0.0f;   // 256 floats: sum[128], sumsq[128]
}

// ---------------- kernel 1: node-level linear transforms (WMMA, M=64 blocking) ----
// a_src[i] = x[i]·Wa[0:128] ; a_dst[i] = x[i]·Wa[128:256]
// xm[i,:]  = x[i]@W_m + b_m                (message, pre-gate)
// h[i,:]   = x[i]@W_r + b_r                (residual; edge kernel atomically adds msgs)
__global__ __launch_bounds__(256) void k_node_gemm(
    const float* __restrict__ x,
    const float* __restrict__ Wa,
    const float* __restrict__ Wm, const float* __restrict__ bm,
    const float* __restrict__ Wr, const float* __restrict__ br,
    float* __restrict__ xm, float* __restrict__ h,
    float* __restrict__ a_src, float* __restrict__ a_dst, int n)
{
    __shared__ float xt[64 * D];      // 64x128 f32 node tile (32 KB)

    const int t  = threadIdx.x;
    const int m0 = blockIdx.x * 64;

    // ---- stage 64x128 x tile via dwordx4 loads (rows 512B-aligned); clamp tail rows
    for (int f = t; f < 2048; f += 256) {
        int r  = f >> 5;
        int gr = m0 + r; if (gr >= n) gr = n - 1;
        ((float4*)xt)[f] = ((const float4*)(x + (size_t)gr * D))[f & 31];
    }
    __syncthreads();

    // ---- gate-logit projections: node r = t>>2, quarter q = t&3 (32 k each),
    //      reduce across 4 adjacent lanes with wave shuffles
    {
        int r = t >> 2, q = t & 3;
        float ps = 0.f, pd = 0.f;
        #pragma unroll
        for (int j = 0; j < 32; ++j) {
            float xv = xt[r * D + q * 32 + j];
            ps += xv * Wa[q * 32 + j];
            pd += xv * Wa[D + q * 32 + j];
        }
        ps += __shfl_xor(ps, 1); ps += __shfl_xor(ps, 2);
        pd += __shfl_xor(pd, 1); pd += __shfl_xor(pd, 2);
        int gr = m0 + r;
        if (q == 0 && gr < n) { a_src[gr] = ps; a_dst[gr] = pd; }
    }

    // ---- WMMA: 8 waves; wave w owns 32 cols of [W_m | W_r] (waves 0-3 -> xm, 4-7 -> h)
    //      4 M-tiles per wave reuse each B fragment 4x.
    const int wave = t >> 5;
    const int lane = t & 31;
    const int row  = lane & 15;                 // A-matrix M row / C-matrix N col
    const int hi   = lane >> 4;                 // 0: lanes 0-15, 1: lanes 16-31
    const float* W   = (wave < 4) ? Wm : Wr;
    const float* Bv  = (wave < 4) ? bm : br;
    float*       out = (wave < 4) ? xm : h;
    const int cb   = (wave & 3) * 32;
    const int col0 = cb + row;
    const int col1 = col0 + 16;

    v8f acc[4][2];
    {   // seed accumulators with bias (bias depends only on N-column)
        float b0 = Bv[col0], b1 = Bv[col1];
        #pragma unroll
        for (int mt = 0; mt < 4; ++mt)
            #pragma unroll
            for (int j = 0; j < 8; ++j) { acc[mt][0][j] = b0; acc[mt][1][j] = b1; }
    }

#if defined(__gfx1250__)
    // f32 WMMA: K=4 per instruction, 32 steps over K=128.
    // A 16x4 layout: lanes0-15 VGPR{0,1}=K{0,1}; lanes16-31 VGPR{0,1}=K{2,3}
    // B 4x16 layout: lanes0-15 rows K{0,1}; lanes16-31 rows K{2,3}; N = lane&15
    const int khalf = hi * 2;
    #pragma unroll 2
    for (int ks = 0; ks < 32; ++ks) {
        const int k = ks * 4 + khalf;
        v2f b0, b1;
        b0.x = W[(size_t)k * D + col0];  b0.y = W[(size_t)(k + 1) * D + col0];
        b1.x = W[(size_t)k * D + col1];  b1.y = W[(size_t)(k + 1) * D + col1];
        #pragma unroll
        for (int mt = 0; mt < 4; ++mt) {
            const float* xr = xt + (mt * 16 + row) * D + k;
            v2f a; a.x = xr[0]; a.y = xr[1];
            acc[mt][0] = __builtin_amdgcn_wmma_f32_16x16x4_f32(false, a, false, b0,
                                                              (short)0, acc[mt][0], false, false);
            acc[mt][1] = __builtin_amdgcn_wmma_f32_16x16x4_f32(false, a, false, b1,
                                                              (short)0, acc[mt][1], false, false);
        }
    }
#else
    // host-compile parse body (never codegen'd for device): scalar reference
    for (int k = 0; k < D; ++k) {
        #pragma unroll
        for (int mt = 0; mt < 4; ++mt) {
            float av = xt[(mt * 16 + row) * D + k];
            #pragma unroll
            for (int j = 0; j < 8; ++j) {
                acc[mt][0][j] += av * W[(size_t)k * D + col0];
                acc[mt][1][j] += av * W[(size_t)k * D + col1];
            }
        }
    }
#endif

    // C/D layout: VGPR j -> M = j + 8*hi, N = lane&15
    const int mb = hi * 8;
    if (m0 + 64 <= n) {                       // uniform branch: full block, branchless stores
        #pragma unroll
        for (int mt = 0; mt < 4; ++mt) {
            #pragma unroll
            for (int j = 0; j < 8; ++j) {
                size_t gr = (size_t)(m0 + mt * 16 + mb + j);
                out[gr * D + col0] = acc[mt][0][j];
                out[gr * D + col1] = acc[mt][1][j];
            }
        }
    } else {                                  // ragged tail block only
        #pragma unroll
        for (int mt = 0; mt < 4; ++mt) {
            #pragma unroll
            for (int j = 0; j < 8; ++j) {
                int gr = m0 + mt * 16 + mb + j;
                if (gr < n) {
                    out[(size_t)gr * D + col0] = acc[mt][0][j];
                    out[(size_t)gr * D + col1] = acc[mt][1][j];
                }
            }
        }
    }
}

// ---------------- kernel 2: gated edge scatter (one wave per edge) ----------------
__global__ __launch_bounds__(256) void k_edge_scatter(
    const int* __restrict__ src, const int* __restrict__ dst,
    const float* __restrict__ a_src, const float* __restrict__ a_dst,
    const float* __restrict__ ba,
    const float* __restrict__ xm, float* __restrict__ h, int e)
{
    const int lane   = threadIdx.x & 31;
    const int wid    = (blockIdx.x * blockDim.x + threadIdx.x) >> 5;
    const int nwaves = (gridDim.x * blockDim.x) >> 5;
    const float bav  = ba[0];

    for (int i = wid; i < e; i += nwaves) {
        int s = src[i];
        int d = dst[i];
        float z = a_src[s] + a_dst[d] + bav;
        float g = 1.0f / (1.0f + __expf(-z));                 // sigmoid gate (wave-uniform)
        float4 v = *(const float4*)(xm + (size_t)s * D + lane * 4);   // L2-resident gather
        float* hp = h + (size_t)d * D + lane * 4;
        atomicAdd(hp + 0, g * v.x);                            // global_atomic_add_f32 (L2)
        atomicAdd(hp + 1, g * v.y);
        atomicAdd(hp + 2, g * v.z);
        atomicAdd(hp + 3, g * v.w);
    }
}

// ---------------- kernel 3: per-column sum / sumsq ----------------
__global__ __launch_bounds__(128) void k_col_stats(
    const float* __restrict__ h, float* __restrict__ stats, int n)
{
    const int c = threadIdx.x;   // 128 columns
    float s = 0.f, s2 = 0.f;
    for (int r = blockIdx.x; r < n; r += gridDim.x) {
        float v = h[(size_t)r * D + c];
        s += v; s2 += v * v;
    }
    atomicAdd(&stats[c], s);
    atomicAdd(&stats[D + c], s2);
}

// ---------------- kernel 4: batchnorm + relu ----------------
__global__ __launch_bounds__(256) void k_bn_relu(
    const float* __restrict__ h, const float* __restrict__ stats,
    const float* __restrict__ gamma, const float* __restrict__ beta,
    float* __restrict__ out, int n)
{
    const int total = n * D;
    int i = blockIdx.x * blockDim.x + threadIdx.x;
    if (i >= total) return;
    int c = i & (D - 1);
    float inv_n = 1.0f / (float)n;
    float mu  = stats[c] * inv_n;
    float var = stats[D + c] * inv_n - mu * mu;   // E[h^2] - mu^2
    float rs  = rsqrtf(var + 1e-5f);
    float v   = (h[i] - mu) * rs * gamma[c] + beta[c];
    out[i] = v > 0.f ? v : 0.f;
}

// ---------------- host launcher ----------------
extern "C" void kernel_launch(void* const* d_in, const int* in_sizes, int n_in,
                              void* d_out, int out_size, void* d_ws, size_t ws_size,
                              hipStream_t stream)
{
    const float* x    = (const float*)d_in[0];
    const int*   ei   = (const int*)d_in[1];       // edge_index (2,E) flat
    const float* Wa   = (const float*)d_in[2];
    const float* ba   = (const float*)d_in[3];
    const float* Wm   = (const float*)d_in[4];
    const float* bm   = (const float*)d_in[5];
    const float* Wr   = (const float*)d_in[6];
    const float* br   = (const float*)d_in[7];
    const float* gam  = (const float*)d_in[8];
    const float* bet  = (const float*)d_in[9];
    float*       out  = (float*)d_out;

    const int n = in_sizes[0] / D;     // 100000
    const int e = in_sizes[1] / 2;     // 600000
    const int* src = ei;
    const int* dst = ei + e;

    // workspace layout (floats): xm[N*D] | h[N*D] | a_src[N] | a_dst[N] | stats[256]
    float* ws    = (float*)d_ws;
    float* xm    = ws;
    float* h     = ws + (size_t)n * D;
    float* asrc  = ws + 2 * (size_t)n * D;
    float* adst  = asrc + n;
    float* stats = adst + n;

    k_zero_stats<<<1, 256, 0, stream>>>(stats);

    int nb1 = (n + 63) / 64;
    k_node_gemm<<<nb1, 256, 0, stream>>>(x, Wa, Wm, bm, Wr, br, xm, h, asrc, adst, n);

    k_edge_scatter<<<1024, 256, 0, stream>>>(src, dst, asrc, adst, ba, xm, h, e);

    k_col_stats<<<512, 128, 0, stream>>>(h, stats, n);

    int nb4 = (n * D + 255) / 256;
    k_bn_relu<<<nb4, 256, 0, stream>>>(h, stats, gam, bet, out, n);
}